// AverageLayer1_24790551232661
// MI455X (gfx1250) — compile-verified
//
#include <hip/hip_runtime.h>

// 17-tap box filter along seq dim as a banded fp32 WMMA matmul (gfx1250).
// out[b,t,d] = (1/17) * sum_{i=-8..8} x[b,t+i,d]  (zero padded)
//
// OUT(16t x 16d) = sum_{kb=2..9} COEF(16x4) * X(4x16); inputs span rows
// [t0-16, t0+32) = {prev, cur, next} 16-row tiles rotated in registers along
// a 256-row strip. Memory-optimal streaming (1R + 1W + 12.5% strip halo);
// 8 x V_WMMA_F32_16X16X4_F32 per 1KB output tile.

typedef __attribute__((ext_vector_type(2))) float v2f;
typedef __attribute__((ext_vector_type(8))) float v8f;

namespace {
constexpr int kB = 8;
constexpr int kS = 8192;
constexpr int kD = 1024;
constexpr int kStrip = 256;          // t-rows per wave
constexpr int kTiles = kStrip / 16;  // 16 output tiles per strip
constexpr int kDT = kD / 16;         // 64 d-tiles
constexpr int kST = kS / kStrip;     // 32 strips
constexpr int kWavesPerBlock = 8;    // 256 threads, wave32
}

struct Tile { v2f c[4]; };  // four K=4 B-operand chunks of a 16x16 fp32 tile

__global__ __launch_bounds__(256)
void box17_wmma_kernel(const float* __restrict__ x, float* __restrict__ y)
{
    const int lane = threadIdx.x & 31;
    // Wave id is uniform across the wave: force it scalar so the strip /
    // boundary logic compiles to SALU branches (no EXEC mask churn).
    const int wid = __builtin_amdgcn_readfirstlane(
        (int)(blockIdx.x * kWavesPerBlock + (threadIdx.x >> 5)));

    const int st = wid % kST;               // strip index (scalar)
    const int dt = (wid / kST) % kDT;       // d-tile index (scalar)
    const int b  = wid / (kST * kDT);       // batch index (scalar)

    const int m     = lane & 15;            // A row / D column
    const int koff  = (lane >> 4) << 1;     // B/K sub-rows {0,1} vs {2,3}
    const int rhalf = (lane >> 4) << 3;     // D rows r vs r+8

    // Constant banded coefficients, K-blocks kb = 2..9 of the 16x48 band:
    // coef[m][s] = 1/17 iff m+8 <= s <= m+24.
    const float inv = 1.0f / 17.0f;
    v2f A[8];
#pragma unroll
    for (int i = 0; i < 8; ++i) {
        const int s0 = 4 * (i + 2) + koff;
        A[i].x = (s0     >= m + 8 && s0     <= m + 24) ? inv : 0.0f;
        A[i].y = (s0 + 1 >= m + 8 && s0 + 1 <= m + 24) ? inv : 0.0f;
        // Opaque barrier: keep the coefficients resident in VGPRs instead of
        // letting the compiler rematerialize v_cmp/v_cndmask chains per tile.
        asm volatile("" : "+v"(A[i]));
    }

    const int ts = st * kStrip;             // strip start row (scalar)

    // Per-lane base pointers with lane-dependent parts folded in once; all
    // subsequent accesses use compile-time immediate offsets (< +-8MB).
    const long long base = (long long)b * kS * kD + dt * 16 + m;
    const float* __restrict__ xlb = x + base + (long long)(ts + koff) * kD;
    float* __restrict__ ylb       = y + base + (long long)(ts + rhalf) * kD;

    // Load a 16-row tile whose first row is ts + R (R compile-time).
    auto load_rel = [&](int R) {
        Tile T;
#pragma unroll
        for (int k = 0; k < 4; ++k) {
            const float* p = xlb + (long long)(R + 4 * k) * kD;
            T.c[k].x = p[0];    // K row 4k+koff
            T.c[k].y = p[kD];   // K row 4k+koff+1
        }
        return T;
    };
    auto zero_tile = []() {
        Tile T;
#pragma unroll
        for (int k = 0; k < 4; ++k) { T.c[k].x = 0.0f; T.c[k].y = 0.0f; }
        return T;
    };

    // Only two tiles can ever be out of range: the prev-halo of strip 0 and
    // the next-halo of the last strip. Scalar-guard just those.
    Tile P = (st != 0) ? load_rel(-16) : zero_tile();
    Tile C = load_rel(0);

#pragma unroll
    for (int ti = 0; ti < kTiles; ++ti) {
        Tile N;
        if (ti < kTiles - 1) {
            N = load_rel(16 * ti + 16);                 // always in range
        } else {
            N = (st != kST - 1) ? load_rel(16 * ti + 16) : zero_tile();
        }

        v8f acc = {};
        acc = __builtin_amdgcn_wmma_f32_16x16x4_f32(false, A[0], false, P.c[2], (short)0, acc, false, false);
        acc = __builtin_amdgcn_wmma_f32_16x16x4_f32(false, A[1], false, P.c[3], (short)0, acc, false, false);
        acc = __builtin_amdgcn_wmma_f32_16x16x4_f32(false, A[2], false, C.c[0], (short)0, acc, false, false);
        acc = __builtin_amdgcn_wmma_f32_16x16x4_f32(false, A[3], false, C.c[1], (short)0, acc, false, false);
        acc = __builtin_amdgcn_wmma_f32_16x16x4_f32(false, A[4], false, C.c[2], (short)0, acc, false, false);
        acc = __builtin_amdgcn_wmma_f32_16x16x4_f32(false, A[5], false, C.c[3], (short)0, acc, false, false);
        acc = __builtin_amdgcn_wmma_f32_16x16x4_f32(false, A[6], false, N.c[0], (short)0, acc, false, false);
        acc = __builtin_amdgcn_wmma_f32_16x16x4_f32(false, A[7], false, N.c[1], (short)0, acc, false, false);

        // D layout: VGPR r -> rows ti*16+r / +8 (folded into ylb), column
        // dt*16 + (lane&15). Streaming output -> non-temporal stores with
        // immediate offsets off the single ylb address pair.
#pragma unroll
        for (int r = 0; r < 8; ++r) {
            __builtin_nontemporal_store(acc[r], ylb + (long long)(16 * ti + r) * kD);
        }

        P = C;   // rotate the sliding window of input tiles
        C = N;
    }
}

extern "C" void kernel_launch(void* const* d_in, const int* in_sizes, int n_in,
                              void* d_out, int out_size, void* d_ws, size_t ws_size,
                              hipStream_t stream)
{
    (void)in_sizes; (void)n_in; (void)out_size; (void)d_ws; (void)ws_size;
    const float* x = (const float*)d_in[0];
    float* y = (float*)d_out;

    const int nwaves = kB * kDT * kST;               // 16384 waves
    dim3 grid(nwaves / kWavesPerBlock);              // 2048 blocks
    dim3 block(kWavesPerBlock * 32);                 // 256 threads (8 x wave32)
    box17_wmma_kernel<<<grid, block, 0, stream>>>(x, y);
}